// lstm_NN_baseline_65249143161225
// MI455X (gfx1250) — compile-verified
//
#include <hip/hip_runtime.h>
#include <hip/hip_bf16.h>

// ---------------------------------------------------------------------------
// Shapes: T=65536, E=300, H=64, 4H=256, NH=32, O=4
// ---------------------------------------------------------------------------
#define T_STEPS 65536
#define E_DIM   300
#define H_DIM   64
#define G_DIM   256   // 4*H
#define NH_DIM  32
#define O_DIM   4

typedef float v2f __attribute__((ext_vector_type(2)));
typedef float v8f __attribute__((ext_vector_type(8)));

// ---------------------------------------------------------------------------
// Phase 1: x_proj[t, g] = sum_e embeds[t,e] * W_ih[g,e] + b_ih[g] + b_hh[g]
// One wave32 computes one 16x16 tile of x_proj via V_WMMA_F32_16X16X4_F32.
// A-layout (16x4 f32): lanes 0-15 hold M=lane, {K=k,k+1}; lanes 16-31 {K=k+2,k+3}
// B-layout (4x16 f32): vgpr0 = rows K=k (lanes 0-15) / K=k+1 (lanes 16-31),
//                      vgpr1 = rows K=k+2 / K=k+3
// C/D-layout (16x16):  vgpr v, lanes 0-15 -> M=v, lanes 16-31 -> M=v+8, N=lane%16
// ---------------------------------------------------------------------------
__global__ void __launch_bounds__(256)
xproj_wmma_kernel(const float* __restrict__ embeds,   // [T, E]
                  const float* __restrict__ W_ih,     // [4H, E] row-major
                  const float* __restrict__ b_ih,     // [4H]
                  const float* __restrict__ b_hh,     // [4H]
                  float* __restrict__ xproj)          // [T, 4H]
{
    const int lane    = threadIdx.x & 31;
    const int wid     = blockIdx.x * (blockDim.x >> 5) + (threadIdx.x >> 5);
    const int tile_n  = wid & 15;   // 256/16 = 16 column tiles
    const int tile_t  = wid >> 4;   // 65536/16 = 4096 row tiles
    const int half    = lane >> 4;  // 0: lanes 0-15, 1: lanes 16-31
    const int l16     = lane & 15;

    const int arow_i  = tile_t * 16 + l16;   // embeds row this lane feeds (A)
    const int col     = tile_n * 16 + l16;   // output/bias column this lane owns (B, C/D)

    const float* __restrict__ arow = embeds + (size_t)arow_i * E_DIM;
    const float* __restrict__ brow = W_ih   + (size_t)col    * E_DIM;

    v8f acc = {};   // zero accumulator

    // E = 300 = 75 * 4 exactly
    #pragma unroll 5
    for (int k = 0; k < E_DIM; k += 4) {
        // A fragment: 2 consecutive f32 from this lane's embeds row
        const float* ap = arow + k + 2 * half;
        v2f a;
        a[0] = ap[0];
        a[1] = ap[1];

        // B fragment: pick K={k,k+2} (half=0) or K={k+1,k+3} (half=1)
        float4 w = *(const float4*)(brow + k);   // W_ih row is 1200B -> 16B aligned
        v2f b;
        b[0] = half ? w.y : w.x;
        b[1] = half ? w.w : w.z;

        acc = __builtin_amdgcn_wmma_f32_16x16x4_f32(
                  /*neg_a=*/false, a, /*neg_b=*/false, b,
                  /*c_mod=*/(short)0, acc,
                  /*reuse_a=*/false, /*reuse_b=*/false);
    }

    const float bias = b_ih[col] + b_hh[col];
    #pragma unroll
    for (int v = 0; v < 8; ++v) {
        const int m = tile_t * 16 + v + 8 * half;
        xproj[(size_t)m * G_DIM + col] = acc[v] + bias;
    }
}

// ---------------------------------------------------------------------------
// Phase 2: serial LSTM scan + MLP head. Single block, 256 threads (8 waves).
// Thread j owns gate row j: W_hh[j, 0:64] lives in 64 VGPRs. h lives in LDS
// (broadcast b128 reads). c lives in a register of threads 0..63.
// x_proj streams from L2 with a +1-step register prefetch.
// ---------------------------------------------------------------------------
__device__ __forceinline__ float sigf(float x) {
    return 1.0f / (1.0f + __expf(-x));
}
__device__ __forceinline__ float tanhf_fast(float x) {
    float e = __expf(2.0f * x);
    return 1.0f - 2.0f / (e + 1.0f);
}

__global__ void __launch_bounds__(256, 1)
lstm_scan_kernel(const float* __restrict__ xproj,     // [T, 4H]
                 const float* __restrict__ W_hh,      // [4H, H] row-major
                 const float* __restrict__ h0,        // [H]
                 const float* __restrict__ c0,        // [H]
                 const float* __restrict__ W1,        // [NH, 2H]
                 const float* __restrict__ b1,        // [NH]
                 const float* __restrict__ W2,        // [O, NH]
                 const float* __restrict__ b2,        // [O]
                 const int*   __restrict__ event_ix,  // [2]
                 float* __restrict__ out)             // [O]
{
    __shared__ __align__(16) float h_s[H_DIM];
    __shared__ float g_s[G_DIM];
    __shared__ __align__(16) float sel_s[2 * H_DIM];
    __shared__ float hnn_s[NH_DIM];

    const int j = threadIdx.x;

    // Preload this thread's W_hh row into registers (64 VGPRs).
    float w[H_DIM];
    #pragma unroll
    for (int k4 = 0; k4 < 16; ++k4) {
        float4 t = *(const float4*)(W_hh + (size_t)j * H_DIM + 4 * k4);
        w[4 * k4 + 0] = t.x;
        w[4 * k4 + 1] = t.y;
        w[4 * k4 + 2] = t.z;
        w[4 * k4 + 3] = t.w;
    }

    float c = 0.0f;
    if (j < H_DIM) {
        h_s[j] = h0[j];
        c      = c0[j];
    }
    const int ev0 = event_ix[0];
    const int ev1 = event_ix[1];
    __syncthreads();

    float xg = xproj[j];   // t = 0 pre-gate input

    for (int t = 0; t < T_STEPS; ++t) {
        // Prefetch next step's pre-gate value (hits L2; latency hides under matvec)
        const int tn = (t < T_STEPS - 1) ? (t + 1) : (T_STEPS - 1);
        const float xg_next = xproj[(size_t)tn * G_DIM + j];

        // g[j] = xg[j] + h . W_hh[j,:]
        float g = xg;
        const float4* h4 = (const float4*)h_s;
        #pragma unroll
        for (int k4 = 0; k4 < 16; ++k4) {
            float4 hv = h4[k4];            // ds_load_b128, broadcast across lanes
            g = fmaf(hv.x, w[4 * k4 + 0], g);
            g = fmaf(hv.y, w[4 * k4 + 1], g);
            g = fmaf(hv.z, w[4 * k4 + 2], g);
            g = fmaf(hv.w, w[4 * k4 + 3], g);
        }
        g_s[j] = g;
        __syncthreads();   // g ready; also: everyone done reading h_s

        if (j < H_DIM) {
            const float gi = g_s[j];
            const float gf = g_s[H_DIM + j];
            const float gg = g_s[2 * H_DIM + j];
            const float go = g_s[3 * H_DIM + j];
            c = sigf(gf) * c + sigf(gi) * tanhf_fast(gg);
            const float h = sigf(go) * tanhf_fast(c);
            h_s[j] = h;
            if (t == ev0) sel_s[j] = h;
            if (t == ev1) sel_s[H_DIM + j] = h;
        }
        xg = xg_next;
        __syncthreads();   // h ready for next step
    }

    // MLP head: h_nn = relu(flat @ W1^T + b1); out = h_nn @ W2^T + b2
    if (j < NH_DIM) {
        float a = b1[j];
        #pragma unroll 8
        for (int k = 0; k < 2 * H_DIM; ++k)
            a = fmaf(sel_s[k], W1[(size_t)j * 2 * H_DIM + k], a);
        hnn_s[j] = fmaxf(a, 0.0f);
    }
    __syncthreads();
    if (j < O_DIM) {
        float a = b2[j];
        #pragma unroll
        for (int k = 0; k < NH_DIM; ++k)
            a = fmaf(hnn_s[k], W2[(size_t)j * NH_DIM + k], a);
        out[j] = a;
    }
}

// ---------------------------------------------------------------------------
// Launch
// ---------------------------------------------------------------------------
extern "C" void kernel_launch(void* const* d_in, const int* in_sizes, int n_in,
                              void* d_out, int out_size, void* d_ws, size_t ws_size,
                              hipStream_t stream) {
    const float* embeds   = (const float*)d_in[0];
    const float* h0       = (const float*)d_in[1];
    const float* c0       = (const float*)d_in[2];
    const float* W_ih     = (const float*)d_in[3];
    const float* W_hh     = (const float*)d_in[4];
    const float* b_ih     = (const float*)d_in[5];
    const float* b_hh     = (const float*)d_in[6];
    const float* W1       = (const float*)d_in[7];
    const float* b1       = (const float*)d_in[8];
    const float* W2       = (const float*)d_in[9];
    const float* b2       = (const float*)d_in[10];
    const int*   event_ix = (const int*)d_in[11];

    float* xproj = (float*)d_ws;               // needs T*4H*4 = 64 MB of scratch
    float* out   = (float*)d_out;

    // 65536 wave-tiles (4096 t-tiles x 16 n-tiles), 8 waves per 256-thread block
    xproj_wmma_kernel<<<8192, 256, 0, stream>>>(embeds, W_ih, b_ih, b_hh, xproj);

    // Single-workgroup sequential scan + head
    lstm_scan_kernel<<<1, 256, 0, stream>>>(xproj, W_hh, h0, c0,
                                            W1, b1, W2, b2, event_ix, out);
}